// Object_Attention_module_50766513439204
// MI455X (gfx1250) — compile-verified
//
#include <hip/hip_runtime.h>
#include <hip/hip_bf16.h>
#include <math.h>

typedef __attribute__((ext_vector_type(2))) float v2f;
typedef __attribute__((ext_vector_type(8))) float v8f;

#define GEMM_WAVES 8

// Y[n, o] = sum_d X[n, d] * W[o, d] + bias[o]   (n in [0,32), row-major buffers)
// One 16-row M tile of W per workgroup; 8 waves split K; D must be divisible by 256.
__global__ __launch_bounds__(GEMM_WAVES * 32)
void gemm_wmma_kernel(const float* __restrict__ W, const float* __restrict__ X,
                      const float* __restrict__ bias, float* __restrict__ Y,
                      int O, int D) {
    __shared__ float lds[GEMM_WAVES * 16 * 32];

    const int tile = blockIdx.x;           // M tile index
    const int lane = threadIdx.x & 31;
    const int wave = threadIdx.x >> 5;
    const int m16  = lane & 15;
    const int hi   = lane >> 4;            // 0: lanes 0-15, 1: lanes 16-31

    const int kchunk = D / GEMM_WAVES;     // divisible by 32
    const int k0     = wave * kchunk;

    const float* wrow = W + (size_t)(tile * 16 + m16) * D;   // A frag row (M = m16)
    const float* xr0  = X + (size_t)(m16) * D;               // B frag, N tile 0
    const float* xr1  = X + (size_t)(16 + m16) * D;          // B frag, N tile 1

    v8f c0 = {};   // D tile, N = 0..15
    v8f c1 = {};   // D tile, N = 16..31

    for (int k = k0; k < k0 + kchunk; k += 32) {
        // stream-ahead prefetch of the W row (one cacheline per 32 K)
        __builtin_prefetch(wrow + k + 1024, 0, 1);
#pragma unroll
        for (int kk = 0; kk < 32; kk += 4) {
            const int kb = k + kk;
            // A 16x4 f32: lane holds K = hi*2 + {0,1}  (contiguous b64)
            v2f a  = *(const v2f*)(wrow + kb + hi * 2);
            // B 4x16 f32: VGPR j holds K = j + hi*2, N = lane&15 (contiguous b64)
            v2f b0 = *(const v2f*)(xr0 + kb + hi * 2);
            v2f b1 = *(const v2f*)(xr1 + kb + hi * 2);
            c0 = __builtin_amdgcn_wmma_f32_16x16x4_f32(false, a, false, b0,
                                                       (short)0, c0, false, false);
            c1 = __builtin_amdgcn_wmma_f32_16x16x4_f32(false, a, false, b1,
                                                       (short)0, c1, false, false);
        }
    }

    // Scatter this wave's 16x32 partial into LDS.
    // C layout: VGPR j -> M = j + hi*8, N = (lane&15) [+16 for c1]
#pragma unroll
    for (int j = 0; j < 8; ++j) {
        const int m = j + hi * 8;
        lds[(wave * 16 + m) * 32 + m16]      = c0[j];
        lds[(wave * 16 + m) * 32 + 16 + m16] = c1[j];
    }
    __syncthreads();

    // Reduce 8 partials, add bias, store: 512 outputs over 256 threads.
    for (int idx = threadIdx.x; idx < 16 * 32; idx += blockDim.x) {
        const int m = idx >> 5;
        const int n = idx & 31;
        float s = 0.f;
#pragma unroll
        for (int w = 0; w < GEMM_WAVES; ++w) s += lds[(w * 16 + m) * 32 + n];
        const int o = tile * 16 + m;
        Y[(size_t)n * O + o] = s + bias[o];
    }
}

// Training-mode BatchNorm1d over axis 0 (32 rows), in place. One thread per column.
__global__ void bn_kernel(float* __restrict__ buf, const float* __restrict__ gamma,
                          const float* __restrict__ beta, int O) {
    const int o = blockIdx.x * blockDim.x + threadIdx.x;
    if (o >= O) return;
    float vals[32];
    float mu = 0.f;
#pragma unroll
    for (int b = 0; b < 32; ++b) { vals[b] = buf[(size_t)b * O + o]; mu += vals[b]; }
    mu *= (1.f / 32.f);
    float var = 0.f;
#pragma unroll
    for (int b = 0; b < 32; ++b) { float d = vals[b] - mu; var += d * d; }
    var *= (1.f / 32.f);
    const float inv = rsqrtf(var + 1e-5f);
    const float g = gamma[o] * inv, be = beta[o];
#pragma unroll
    for (int b = 0; b < 32; ++b) buf[(size_t)b * O + o] = g * (vals[b] - mu) + be;
}

// Per-(b,h) channel attention: q(32,17), k(32,9), v(32,9) -> out(32,17).
// One wave per (b,h); 256 blocks total.
__global__ __launch_bounds__(32)
void attn_kernel(const float* __restrict__ q, const float* __restrict__ k,
                 const float* __restrict__ v, const float* __restrict__ temp,
                 float* __restrict__ out) {
    __shared__ float qs[32 * 17];
    __shared__ float ks[32 * 9];
    __shared__ float vs[32 * 9];
    __shared__ float as[17 * 9];

    const int b = blockIdx.x >> 3;
    const int h = blockIdx.x & 7;
    const int lane = threadIdx.x;

    const float* qb = q + (size_t)b * 4352 + h * 544;  // 32*17
    const float* kb = k + (size_t)b * 2304 + h * 288;  // 32*9
    const float* vb = v + (size_t)b * 2304 + h * 288;

    for (int i = lane; i < 544; i += 32) qs[i] = qb[i];
    for (int i = lane; i < 288; i += 32) { ks[i] = kb[i]; vs[i] = vb[i]; }
    __syncthreads();

    const float t = temp[h];
    // attn[d,e] = t * sum_cc q[cc,d] * k[cc,e]
    for (int i = lane; i < 153; i += 32) {
        const int d = i / 9, e = i - d * 9;
        float s = 0.f;
#pragma unroll
        for (int cc = 0; cc < 32; ++cc) s += qs[cc * 17 + d] * ks[cc * 9 + e];
        as[i] = s * t;
    }
    __syncthreads();

    // softmax over e (9) per row d (17)
    if (lane < 17) {
        float mx = as[lane * 9];
#pragma unroll
        for (int e = 1; e < 9; ++e) mx = fmaxf(mx, as[lane * 9 + e]);
        float sum = 0.f;
#pragma unroll
        for (int e = 0; e < 9; ++e) {
            const float ex = __expf(as[lane * 9 + e] - mx);
            as[lane * 9 + e] = ex;
            sum += ex;
        }
        const float r = 1.f / sum;
#pragma unroll
        for (int e = 0; e < 9; ++e) as[lane * 9 + e] *= r;
    }
    __syncthreads();

    // out[cc,d] = sum_e attn[d,e] * v[cc,e]
    float* ob = out + (size_t)b * 4352 + h * 544;
    const int cc = lane;
#pragma unroll
    for (int d = 0; d < 17; ++d) {
        float s = 0.f;
#pragma unroll
        for (int e = 0; e < 9; ++e) s += as[d * 9 + e] * vs[cc * 9 + e];
        ob[cc * 17 + d] = s;
    }
}

extern "C" void kernel_launch(void* const* d_in, const int* in_sizes, int n_in,
                              void* d_out, int out_size, void* d_ws, size_t ws_size,
                              hipStream_t stream) {
    (void)in_sizes; (void)n_in; (void)out_size; (void)ws_size;

    const float* x1   = (const float*)d_in[0];   // (32, 18432)
    const float* x2   = (const float*)d_in[1];   // (32, 34816)
    const float* Wk   = (const float*)d_in[2];   // (2304, 18432)
    const float* bk   = (const float*)d_in[3];
    const float* Wv   = (const float*)d_in[4];   // (2304, 18432)
    const float* bv   = (const float*)d_in[5];
    const float* Wq   = (const float*)d_in[6];   // (4352, 34816)
    const float* bq   = (const float*)d_in[7];
    const float* g1   = (const float*)d_in[8];
    const float* be1  = (const float*)d_in[9];
    const float* g2   = (const float*)d_in[10];
    const float* be2  = (const float*)d_in[11];
    const float* g3   = (const float*)d_in[12];
    const float* be3  = (const float*)d_in[13];
    const float* temp = (const float*)d_in[14];  // (8,)
    const float* Wp   = (const float*)d_in[15];  // (4352, 4352)
    const float* bp   = (const float*)d_in[16];
    float* out        = (float*)d_out;           // (32, 4352)

    const int DK = 18432, DQ = 34816, OK = 2304, OQ = 4352;

    float* ws   = (float*)d_ws;
    float* kbuf = ws;                            // 32*2304
    float* vbuf = kbuf + 32 * OK;                // 32*2304
    float* qbuf = vbuf + 32 * OK;                // 32*4352
    float* abuf = qbuf + 32 * OQ;                // 32*4352

    const dim3 blk(GEMM_WAVES * 32);

    // Projections: Y = X @ W^T + b via fp32 WMMA
    gemm_wmma_kernel<<<dim3(OK / 16), blk, 0, stream>>>(Wk, x1, bk, kbuf, OK, DK);
    gemm_wmma_kernel<<<dim3(OK / 16), blk, 0, stream>>>(Wv, x1, bv, vbuf, OK, DK);
    gemm_wmma_kernel<<<dim3(OQ / 16), blk, 0, stream>>>(Wq, x2, bq, qbuf, OQ, DQ);

    // BatchNorm (training stats over 32 rows), in place
    bn_kernel<<<dim3((OK + 255) / 256), dim3(256), 0, stream>>>(kbuf, g1, be1, OK);
    bn_kernel<<<dim3((OK + 255) / 256), dim3(256), 0, stream>>>(vbuf, g2, be2, OK);
    bn_kernel<<<dim3((OQ + 255) / 256), dim3(256), 0, stream>>>(qbuf, g3, be3, OQ);

    // Channel attention per (b, h)
    attn_kernel<<<dim3(32 * 8), dim3(32), 0, stream>>>(qbuf, kbuf, vbuf, temp, abuf);

    // Final projection into d_out
    gemm_wmma_kernel<<<dim3(OQ / 16), blk, 0, stream>>>(Wp, abuf, bp, out, OQ, OQ);
}